// Codec_10625749090429
// MI455X (gfx1250) — compile-verified
//
#include <hip/hip_runtime.h>
#include <hip/hip_bf16.h>

typedef __attribute__((ext_vector_type(8))) float v8f;
typedef __attribute__((ext_vector_type(2))) float v2f;

#define IMG_H 512
#define IMG_W 512
#define STRIP_ROWS 16
#define STRIPS (IMG_H / STRIP_ROWS)   // 32
#define NBINS 256
#define NREP 4
#define HSTRIDE 257                    // pad to rotate LDS banks across replicas

// ---------------------------------------------------------------------------
// Kernel 0: zero the global histogram workspace and the output scalar.
// ---------------------------------------------------------------------------
__global__ void codec_zero(float* __restrict__ gHist, float* __restrict__ out, int n) {
    int i = blockIdx.x * blockDim.x + threadIdx.x;
    if (i < n) gHist[i] = 0.0f;
    if (i == 0) out[0] = 0.0f;
}

// ---------------------------------------------------------------------------
// Kernel 1: fused color transform + fmod wrap + clamped-grad predictor +
// soft histogram (LDS ds_add_f32, 4 replicas/channel) -> global atomic flush.
// One block = 16-row strip of one image (all 3 channels). 256 threads.
// ---------------------------------------------------------------------------
__global__ __launch_bounds__(256) void codec_hist(const float* __restrict__ x,
                                                  const float* __restrict__ params,
                                                  float* __restrict__ gHist) {
    __shared__ float ybuf[2][3][IMG_W];
    __shared__ float hist[3 * NREP * HSTRIDE];

    const int tid   = threadIdx.x;
    const int img   = blockIdx.x >> 5;     // / STRIPS
    const int strip = blockIdx.x & 31;     // % STRIPS
    const int row0  = strip * STRIP_ROWS;

    float p[12];
#pragma unroll
    for (int i = 0; i < 12; ++i) p[i] = params[i];

    for (int i = tid; i < 3 * NREP * HSTRIDE; i += 256) hist[i] = 0.0f;

    const size_t plane = (size_t)IMG_H * IMG_W;
    const float* xr = x + (size_t)(img * 3 + 0) * plane;
    const float* xg = x + (size_t)(img * 3 + 1) * plane;
    const float* xb = x + (size_t)(img * 3 + 2) * plane;

    // exact fmod(t, 2) for |t| < 4
    auto wrap = [](float v) -> float {
        float t = v + 1.0f;
        float m = t - truncf(t * 0.5f) * 2.0f;
        return m - 1.0f;
    };

    // compute transformed row h into ybuf[buf]
    auto transform_row = [&](int h, int buf) {
#pragma unroll
        for (int k = 0; k < 2; ++k) {
            int col = tid + k * 256;
            int idx = h * IMG_W + col;
            // prefetch next row (one lane per 128B cache line)
            if (((col & 31) == 0) && (h + 1 < IMG_H)) {
                __builtin_prefetch(xr + idx + IMG_W, 0, 1);
                __builtin_prefetch(xg + idx + IMG_W, 0, 1);
                __builtin_prefetch(xb + idx + IMG_W, 0, 1);
            }
            float r0 = xr[idx], g0 = xg[idx], b0 = xb[idx];
            float r1 = r0 + (p[0] * g0 + p[1] * b0);
            float g1 = g0 + (p[2] * r1 + p[3] * b0);
            float b1 = b0 + (p[4] * r1 + p[5] * g1);
            float r2 = r1 + (p[6] * g1 + p[7] * b1);
            float g2 = g1 + (p[8] * r2 + p[9] * b1);
            float b2 = b1 + (p[10] * r2 + p[11] * g2);
            ybuf[buf][0][col] = wrap(r2);
            ybuf[buf][1][col] = wrap(g2);
            ybuf[buf][2][col] = wrap(b2);
        }
    };

    int prev = 0;
    if (row0 == 0) {
        for (int i = tid; i < 3 * IMG_W; i += 256) (&ybuf[0][0][0])[i] = 0.0f;  // top padding
    } else {
        transform_row(row0 - 1, 0);  // halo row
    }
    __syncthreads();

    const int rep = (tid >> 5) & (NREP - 1);  // replica per wave

    for (int h = row0; h < row0 + STRIP_ROWS; ++h) {
        int cur = prev ^ 1;
        transform_row(h, cur);
        __syncthreads();

#pragma unroll
        for (int k = 0; k < 2; ++k) {
            int col = tid + k * 256;
#pragma unroll
            for (int c = 0; c < 3; ++c) {
                float v  = ybuf[cur][c][col];
                float n  = ybuf[prev][c][col];
                float w_ = (col > 0) ? ybuf[cur][c][col - 1]  : 0.0f;
                float nw = (col > 0) ? ybuf[prev][c][col - 1] : 0.0f;
                float lo = fminf(n, w_);
                float hi = fmaxf(n, w_);
                float pr = fminf(fmaxf(n + w_ - nw, lo), hi);
                float res = v - pr;

                float xs = (res + 1.0f) * 127.5f;
                float fl = floorf(xs);
                float fr = xs - fl;
                int i0 = (int)fl;
                float* hb = &hist[(c * NREP + rep) * HSTRIDE];
                if ((unsigned)i0 < (unsigned)NBINS)       atomicAdd(&hb[i0],     1.0f - fr);
                if ((unsigned)(i0 + 1) < (unsigned)NBINS) atomicAdd(&hb[i0 + 1], fr);
            }
        }
        __syncthreads();
        prev = cur;
    }

    // flush LDS histogram (sum replicas) to global with native f32 atomics
    for (int t = tid; t < 3 * NBINS; t += 256) {
        int c = t >> 8, bin = t & 255;
        float s = hist[(c * NREP + 0) * HSTRIDE + bin]
                + hist[(c * NREP + 1) * HSTRIDE + bin]
                + hist[(c * NREP + 2) * HSTRIDE + bin]
                + hist[(c * NREP + 3) * HSTRIDE + bin];
        unsafeAtomicAdd(&gHist[((size_t)img * 3 + c) * NBINS + bin], s);
    }
}

// ---------------------------------------------------------------------------
// Kernel 2: entropy of each 256-bin histogram, summed with
// V_WMMA_F32_16X16X4_F32 (A = 16x4 chunk of terms, B = ones), mean/8 -> out.
// One wave (32 lanes) per image; EXEC all ones at the WMMA.
// ---------------------------------------------------------------------------
__global__ __launch_bounds__(32) void codec_entropy_wmma(const float* __restrict__ gHist,
                                                         float* __restrict__ out) {
    __shared__ float terms[NBINS];
    const int lane = threadIdx.x;
    const int img  = blockIdx.x;  // 0..95
    const float invN = 1.0f / (float)(IMG_H * IMG_W);

#pragma unroll
    for (int j = 0; j < 8; ++j) {
        int bin = lane + 32 * j;
        float pp = gHist[(size_t)img * NBINS + bin] * invN;
        terms[bin] = (pp > 0.0f) ? (-pp * log2f(pp)) : 0.0f;
    }
    __syncthreads();

    // A(m,k) = terms[chunk*64 + k*16 + m]; ISA A layout:
    // lanes 0-15: M=lane, VGPR0=K0, VGPR1=K1; lanes 16-31: M=lane-16, K2/K3.
    v8f acc = {};
    v2f ones;
    ones[0] = 1.0f; ones[1] = 1.0f;
#pragma unroll
    for (int c = 0; c < 4; ++c) {
        int base = c * 64 + ((lane & 16) << 1) + (lane & 15);
        v2f a;
        a[0] = terms[base];
        a[1] = terms[base + 16];
        acc = __builtin_amdgcn_wmma_f32_16x16x4_f32(
            /*neg_a=*/false, a, /*neg_b=*/false, ones,
            /*c_mod=*/(short)0, acc, /*reuse_a=*/false, /*reuse_b=*/false);
    }
    // D(m,n) = rowsum_m (identical for every n). Lanes 0-15 hold M=0..7 in
    // VGPR0..7, lanes 16-31 hold M=8..15 -> pair-reduce across half-waves.
    float s = acc[0] + acc[1] + acc[2] + acc[3] + acc[4] + acc[5] + acc[6] + acc[7];
    s += __shfl_xor(s, 16, 32);
    if (lane == 0) unsafeAtomicAdd(out, s * (1.0f / (96.0f * 8.0f)));
}

// ---------------------------------------------------------------------------
extern "C" void kernel_launch(void* const* d_in, const int* in_sizes, int n_in,
                              void* d_out, int out_size, void* d_ws, size_t ws_size,
                              hipStream_t stream) {
    (void)in_sizes; (void)n_in; (void)out_size; (void)ws_size;
    const float* x      = (const float*)d_in[0];  // (32,3,512,512) f32
    const float* params = (const float*)d_in[1];  // (12,) f32
    float* out   = (float*)d_out;                 // scalar f32
    float* gHist = (float*)d_ws;                  // 96*256 f32 = 96 KiB

    codec_zero<<<96, 256, 0, stream>>>(gHist, out, 96 * NBINS);
    codec_hist<<<dim3(32 * STRIPS), 256, 0, stream>>>(x, params, gHist);
    codec_entropy_wmma<<<96, 32, 0, stream>>>(gHist, out);
}